// DeltaSynapse_6184752906853
// MI455X (gfx1250) — compile-verified
//
#include <hip/hip_runtime.h>
#include <hip/hip_bf16.h>
#include <stdint.h>

// Problem constants (match reference): D=8 delays, B=8 batch, N=2048 neurons.
#define NN_    2048
#define BB_    8
#define DD_    8
#define CHUNKS 32              // e-range split for partial sums
#define EPC    (NN_ / CHUNKS)  // 64 presynaptic e's per chunk

typedef __attribute__((ext_vector_type(2))) float v2f;
typedef __attribute__((ext_vector_type(8))) float v8f;

// ---------------------------------------------------------------------------
// Kernel 1: memory-bound fused pass.
// Thread <-> 4 consecutive output neurons o (float4 loads, b128 vmem).
// Each thread keeps all 8 batch accumulators so batch-shared arrays
// (delaymap, W, frac, signs) are read exactly once from HBM.
// Xd[d,b,e] (shared by every o) is staged into LDS as a 64-bit spike mask
// per e: bit (d*8+b).
// ---------------------------------------------------------------------------
__global__ __launch_bounds__(256)
void delta_synapse_partials(const float* __restrict__ Xd,     // (D,B,N)
                            const float* __restrict__ dmap,   // (D,N,N)
                            const float* __restrict__ W,      // (N,N)
                            const float* __restrict__ Wlong,  // (B,N,N)
                            const float* __restrict__ frac,   // (N,N)
                            const float* __restrict__ signs,  // (N,N)
                            float* __restrict__ P)            // (CHUNKS,B,N)
{
    __shared__ uint32_t spkLo[EPC];
    __shared__ uint32_t spkHi[EPC];

    const int tid   = threadIdx.x;
    const int chunk = blockIdx.y;
    const int e0    = chunk * EPC;
    const int o4    = blockIdx.x * 1024 + tid * 4;  // 4 consecutive o's

    // Stage packed spikes for this e-chunk (coalesced over e for fixed d,b).
    if (tid < EPC) {
        const int e = e0 + tid;
        uint32_t lo = 0u, hi = 0u;
#pragma unroll
        for (int d = 0; d < DD_; ++d) {
#pragma unroll
            for (int b = 0; b < BB_; ++b) {
                const float v = Xd[d * (BB_ * NN_) + b * NN_ + e];
                const uint32_t bit = (v > 0.5f) ? 1u : 0u;
                if (d < 4) lo |= bit << (d * 8 + b);
                else       hi |= bit << ((d - 4) * 8 + b);
            }
        }
        spkLo[tid] = lo;
        spkHi[tid] = hi;
    }
    __syncthreads();

    float acc[BB_][4];
#pragma unroll
    for (int b = 0; b < BB_; ++b)
#pragma unroll
        for (int j = 0; j < 4; ++j) acc[b][j] = 0.0f;

    for (int el = 0; el < EPC; ++el) {
        const int e    = e0 + el;
        const int base = e * NN_ + o4;
        const uint32_t lo = spkLo[el];
        const uint32_t hi = spkHi[el];

        // One-hot delay planes: 8 x b128 loads, contiguous across lanes.
        float4 dm[DD_];
#pragma unroll
        for (int d = 0; d < DD_; ++d)
            dm[d] = *(const float4*)(dmap + (size_t)d * NN_ * NN_ + base);

        const float4 w4 = *(const float4*)(W     + base);
        const float4 f4 = *(const float4*)(frac  + base);
        const float4 s4 = *(const float4*)(signs + base);

        float4 wl[BB_];
#pragma unroll
        for (int b = 0; b < BB_; ++b)
            wl[b] = *(const float4*)(Wlong + (size_t)b * NN_ * NN_ + base);

#pragma unroll
        for (int j = 0; j < 4; ++j) {
            // Recover integer delay index from the one-hot plane (exact).
            float fidx = 0.0f;
#pragma unroll
            for (int d = 1; d < DD_; ++d)
                fidx = fmaf((float)d, ((const float*)&dm[d])[j], fidx);
            const int didx = (int)(fidx + 0.5f);
            const int sh   = didx * 8;
            const uint32_t byte =
                ((didx < 4) ? (lo >> sh) : (hi >> (sh - 32))) & 0xFFu;

            const float wj = ((const float*)&w4)[j];
            const float fj = ((const float*)&f4)[j];
            const float sj = ((const float*)&s4)[j];
            const float a_st = sj * wj * (1.0f - fj);  // signs*W*(1-frac)
            const float fs   = sj * fj;                // signs*frac

#pragma unroll
            for (int b = 0; b < BB_; ++b) {
                const float t = fmaf(fs, ((const float*)&wl[b])[j], a_st);
                acc[b][j] += ((byte >> b) & 1u) ? t : 0.0f;
            }
        }
    }

    // Deterministic partial-sum store (no float atomics -> bit-stable replays).
#pragma unroll
    for (int b = 0; b < BB_; ++b) {
        float4 r = make_float4(acc[b][0], acc[b][1], acc[b][2], acc[b][3]);
        *(float4*)(P + (size_t)(chunk * BB_ + b) * NN_ + o4) = r;
    }
}

// ---------------------------------------------------------------------------
// Kernel 2: chunk reduction via V_WMMA_F32_16X16X4_F32.
// A = all-ones 16x4  =>  D[m,n] = sum_k B[k,n]: a column-sum engine.
// Each wave owns one (batch b, 16-wide o tile); 8 accumulating WMMAs sum
// all 32 chunk partials. With A=ones the result only depends on the
// column<->lane mapping (lane&15 = N), which matches the documented layout.
// ---------------------------------------------------------------------------
__global__ __launch_bounds__(128)
void delta_synapse_reduce_wmma(const float* __restrict__ P,  // (CHUNKS,B,N)
                               float* __restrict__ out)      // (B,N)
{
    const int lane  = threadIdx.x & 31;
    const int wid   = (blockIdx.x * blockDim.x + threadIdx.x) >> 5;
    const int b     = wid >> 7;        // 0..7
    const int otile = wid & 127;       // 0..127
    const int o0    = otile * 16;
    const int half  = lane >> 4;
    const int col   = lane & 15;

    v2f a;
    a.x = 1.0f;
    a.y = 1.0f;
    v8f c = {};

#pragma unroll
    for (int cb = 0; cb < CHUNKS; cb += 4) {
        const int k0 = cb + half * 2;  // two K-rows per lane-half
        v2f bv;
        bv.x = P[(size_t)((k0 + 0) * BB_ + b) * NN_ + o0 + col];
        bv.y = P[(size_t)((k0 + 1) * BB_ + b) * NN_ + o0 + col];
        // D = ones(16x4) x B(4x16) + C  -> every row of D holds column sums
        c = __builtin_amdgcn_wmma_f32_16x16x4_f32(
                /*neg_a=*/false, a, /*neg_b=*/false, bv,
                /*c_mod=*/(short)0, c, /*reuse_a=*/false, /*reuse_b=*/false);
    }

    // Row M=0 lives in D VGPR0, lanes 0..15 (N = lane).
    if (lane < 16) out[b * NN_ + o0 + col] = c[0];
}

extern "C" void kernel_launch(void* const* d_in, const int* in_sizes, int n_in,
                              void* d_out, int out_size, void* d_ws, size_t ws_size,
                              hipStream_t stream) {
    const float* Xd    = (const float*)d_in[0];
    const float* dmap  = (const float*)d_in[1];
    const float* W     = (const float*)d_in[2];
    const float* Wlong = (const float*)d_in[3];
    const float* frac  = (const float*)d_in[4];
    const float* signs = (const float*)d_in[5];
    float* P = (float*)d_ws;  // CHUNKS*B*N floats = 2 MB scratch

    // Kernel 1: grid (o-blocks of 1024, e-chunks). 256 thr * 4 o = 1024 o.
    dim3 g1(NN_ / 1024, CHUNKS);
    delta_synapse_partials<<<g1, 256, 0, stream>>>(Xd, dmap, W, Wlong, frac,
                                                   signs, P);

    // Kernel 2: B * (N/16) = 1024 wave-tasks; 128 thr/block = 4 waves.
    delta_synapse_reduce_wmma<<<(BB_ * (NN_ / 16) * 32) / 128, 128, 0, stream>>>(
        P, (float*)d_out);
}